// PAR_18339510354354
// MI455X (gfx1250) — compile-verified
//
#include <hip/hip_runtime.h>
#include <hip/hip_bf16.h>
#include <cstdint>

// ---- problem constants ----
#define H 128
#define W 128
#define HW (H * W)
#define NB 8
#define NC_IMG 3
#define NC_MSK 21
#define K 32
#define NUM_ITER 10
#define TILE 16
#define HALO 8
#define LW 32            // (TILE + 2*HALO) = 32x32 LDS tile
#define W1 0.3f
#define W2 0.01f

// neighbor offset tables (dy = di-1, dx = dj-1 from the reference OFFS)
#define OFFY(i) ((i) < 3 ? -1 : ((i) < 5 ? 0 : 1))
#define OFFX(i) (((i) == 0 || (i) == 3 || (i) == 5) ? -1 : (((i) == 1 || (i) == 6) ? 0 : 1))

// ------------------------------------------------------------------
// Kernel A: posAdd[k] = W2 * softmax(-(pos/((std+eps)*W1))^2)[k]
// ------------------------------------------------------------------
__global__ void pos_kernel(float* __restrict__ posAdd) {
    const int k = threadIdx.x;                     // 32 threads
    const float SQ2 = 1.41421356237309515f;
    float vals[K];
    float s1 = 0.f, s2 = 0.f;
#pragma unroll
    for (int i = 0; i < K; ++i) {
        const int o = i & 7;
        const float pv = (o == 0 || o == 2 || o == 5 || o == 7) ? SQ2 : 1.0f;
        const float v = pv * (float)(1 << (i >> 3));
        vals[i] = v; s1 += v; s2 += v * v;
    }
    const float mean = s1 * (1.0f / 32.0f);
    const float var  = (s2 - 32.0f * mean * mean) * (1.0f / 31.0f);
    const float sd   = sqrtf(fmaxf(var, 0.0f));
    const float invs = 1.0f / ((sd + 1e-8f) * W1);
    float mx = -1e30f;
#pragma unroll
    for (int i = 0; i < K; ++i) {
        const float t = vals[i] * invs;
        vals[i] = -t * t;
        mx = fmaxf(mx, vals[i]);
    }
    float sum = 0.f;
#pragma unroll
    for (int i = 0; i < K; ++i) { vals[i] = __expf(vals[i] - mx); sum += vals[i]; }
    posAdd[k] = W2 * vals[k] / sum;
}

// ------------------------------------------------------------------
// Kernel B: aff[b][p][k] = softmax_k(pixel affinity) + posAdd[k]
// 16x16 pixel tile per block; all 3 image channels' clamped 32x32
// halo tiles staged into LDS with async global->LDS loads; interior
// tiles use one async b128 per lane per channel.
// ------------------------------------------------------------------
__global__ __launch_bounds__(256) void aff_kernel(const float* __restrict__ imgs,
                                                  const float* __restrict__ posAdd,
                                                  float* __restrict__ aff) {
    __shared__ float tile[NC_IMG][LW * LW];        // 12 KB
    const int tid = threadIdx.x;                   // 0..255
    const int tx = tid & (TILE - 1), ty = tid >> 4;
    const int ty0 = (blockIdx.x >> 3) * TILE;      // 8 tiles per row
    const int tx0 = (blockIdx.x & 7) * TILE;
    const int b = blockIdx.y;
    const int y = ty0 + ty, x = tx0 + tx;
    const int p = (y << 7) + x;
    const float* img = imgs + (size_t)b * NC_IMG * HW;

    const bool interior = (ty0 >= HALO) && (ty0 + TILE + HALO <= H) &&
                          (tx0 >= HALO) && (tx0 + TILE + HALO <= W);
    const uint32_t ldsBase = (uint32_t)(uintptr_t)(&tile[0][0]);

    if (interior) {
        // lane fetches 4 contiguous row elements: one async b128 per channel
        const int e = tid * 4;                     // 0,4,...,1020
        const int ly = e >> 5, lx = e & 31;
        const int gidx = ((ty0 - HALO + ly) << 7) + (tx0 - HALO + lx);
#pragma unroll
        for (int c = 0; c < NC_IMG; ++c) {
            const uint32_t dOff = ldsBase + (uint32_t)(c * (LW * LW) + e) * 4u;
            const uint64_t ga = (uint64_t)(uintptr_t)(img + c * HW + gidx);
            asm volatile("global_load_async_to_lds_b128 %0, %1, off"
                         :: "v"(dOff), "v"(ga) : "memory");
        }
    } else {
#pragma unroll
        for (int t = 0; t < 4; ++t) {
            const int e = tid + t * 256;           // 0..1023
            const int ly = e >> 5, lx = e & 31;
            int gy = ty0 - HALO + ly; gy = gy < 0 ? 0 : (gy > H - 1 ? H - 1 : gy);
            int gx = tx0 - HALO + lx; gx = gx < 0 ? 0 : (gx > W - 1 ? W - 1 : gx);
            const int gidx = (gy << 7) + gx;
#pragma unroll
            for (int c = 0; c < NC_IMG; ++c) {
                const uint32_t dOff = ldsBase + (uint32_t)(c * (LW * LW) + e) * 4u;
                const uint64_t ga = (uint64_t)(uintptr_t)(img + c * HW + gidx);
                asm volatile("global_load_async_to_lds_b32 %0, %1, off"
                             :: "v"(dOff), "v"(ga) : "memory");
            }
        }
    }
    asm volatile("s_wait_asynccnt 0x0" ::: "memory");
    __syncthreads();

    // per-channel std over the K neighbors (ddof=1), accumulate affinity
    float ak[K];
#pragma unroll
    for (int kk = 0; kk < K; ++kk) ak[kk] = 0.f;

#pragma unroll
    for (int c = 0; c < NC_IMG; ++c) {
        const float* bse = &tile[c][0] + ty * LW + tx;     // tile corner for this lane
        const float cen = bse[HALO * LW + HALO];
        float nb[K];
        float s1 = 0.f, s2 = 0.f;
#pragma unroll
        for (int kk = 0; kk < K; ++kk) {
            const int d = 1 << (kk >> 3);
            const int o = kk & 7;
            const int off = (HALO + OFFY(o) * d) * LW + (HALO + OFFX(o) * d); // >= 0
            const float v = bse[off];
            nb[kk] = v; s1 += v; s2 = fmaf(v, v, s2);
        }
        const float var = (s2 - s1 * s1 * (1.0f / 32.0f)) * (1.0f / 31.0f);
        const float sd  = sqrtf(fmaxf(var, 0.0f));
        const float inv = 1.0f / ((sd + 1e-8f) * W1);
#pragma unroll
        for (int kk = 0; kk < K; ++kk) {
            const float dlt = (nb[kk] - cen) * inv;
            ak[kk] = fmaf(dlt, dlt, ak[kk]);
        }
    }

    // softmax over K + position term
    float mx = -1e30f;
#pragma unroll
    for (int kk = 0; kk < K; ++kk) { ak[kk] = -ak[kk] * (1.0f / 3.0f); mx = fmaxf(mx, ak[kk]); }
    float sum = 0.f;
#pragma unroll
    for (int kk = 0; kk < K; ++kk) { ak[kk] = __expf(ak[kk] - mx); sum += ak[kk]; }
    const float is = 1.0f / sum;

    // [b][p][k]: 32 contiguous, 128B-aligned -> merged into b128 stores
    float* ap = aff + ((size_t)b * HW + p) * K;
#pragma unroll
    for (int kk = 0; kk < K; ++kk) ap[kk] = fmaf(ak[kk], is, posAdd[kk]);
}

// ------------------------------------------------------------------
// Kernel C: one propagation iteration.
// 16x16 pixel tile per block; 32x32 halo tile per channel staged into
// LDS with async global->LDS loads, double-buffered across channels.
// ------------------------------------------------------------------
__global__ __launch_bounds__(256) void iter_kernel(const float* __restrict__ src,
                                                   const float* __restrict__ aff,
                                                   float* __restrict__ dst) {
    __shared__ float lds[2][LW * LW];              // 8 KB
    const int tid = threadIdx.x;                   // 0..255
    const int tx = tid & (TILE - 1), ty = tid >> 4;
    const int ty0 = (blockIdx.x >> 3) * TILE;      // 8 tiles per row
    const int tx0 = (blockIdx.x & 7) * TILE;
    const int b = blockIdx.y;
    const int y = ty0 + ty, x = tx0 + tx;
    const int p = (y << 7) + x;

    const bool interior = (ty0 >= HALO) && (ty0 + TILE + HALO <= H) &&
                          (tx0 >= HALO) && (tx0 + TILE + HALO <= W);

    // clamped element indices for the boundary (b32) path
    int gidx[4];
#pragma unroll
    for (int t = 0; t < 4; ++t) {
        const int e = tid + t * 256;               // 0..1023
        const int ly = e >> 5, lx = e & 31;
        int gy = ty0 - HALO + ly; gy = gy < 0 ? 0 : (gy > H - 1 ? H - 1 : gy);
        int gx = tx0 - HALO + lx; gx = gx < 0 ? 0 : (gx > W - 1 ? W - 1 : gx);
        gidx[t] = (gy << 7) + gx;
    }
    // unclamped index for the interior (b128) path: 4 contiguous row elements
    const int e4 = tid * 4;
    const int gidx128 = ((ty0 - HALO + (e4 >> 5)) << 7) + (tx0 - HALO + (e4 & 31));

    // per-pixel 32 weights: contiguous [b][p][k] -> 8x global_load_b128
    const float* affp = aff + ((size_t)b * HW + p) * K;
    float wk[K];
#pragma unroll
    for (int kk = 0; kk < K; ++kk) wk[kk] = affp[kk];

    const float* srcb = src + (size_t)b * NC_MSK * HW;
    float*       dstb = dst + (size_t)b * NC_MSK * HW;

    // low 32 bits of the generic LDS address == byte offset in LDS
    const uint32_t ldsBase = (uint32_t)(uintptr_t)(&lds[0][0]);

    auto issue = [&](int c, int buf) {
        const float* cp = srcb + (size_t)c * HW;
        if (interior) {
            const uint32_t dOff = ldsBase + (uint32_t)(buf * (LW * LW) + e4) * 4u;
            const uint64_t ga = (uint64_t)(uintptr_t)(cp + gidx128);
            asm volatile("global_load_async_to_lds_b128 %0, %1, off"
                         :: "v"(dOff), "v"(ga) : "memory");
        } else {
#pragma unroll
            for (int t = 0; t < 4; ++t) {
                const uint32_t dOff = ldsBase + (uint32_t)(buf * (LW * LW) + tid + t * 256) * 4u;
                const uint64_t ga = (uint64_t)(uintptr_t)(cp + gidx[t]);
                asm volatile("global_load_async_to_lds_b32 %0, %1, off"
                             :: "v"(dOff), "v"(ga) : "memory");
            }
        }
    };

    issue(0, 0);                                   // prologue
    for (int c = 0; c < NC_MSK; ++c) {
        if (c + 1 < NC_MSK) {
            issue(c + 1, (c + 1) & 1);             // keep next channel in flight
            if (interior) asm volatile("s_wait_asynccnt 0x1" ::: "memory");
            else          asm volatile("s_wait_asynccnt 0x4" ::: "memory");
        } else {
            asm volatile("s_wait_asynccnt 0x0" ::: "memory");
        }
        __syncthreads();                           // all waves' tiles for buf[c&1] visible

        // 32-tap spatially-varying filter from LDS (constant ds offsets)
        const float* base = &lds[c & 1][0] + ty * LW + tx;   // tile corner for this lane
        float acc = 0.f;
#pragma unroll
        for (int kk = 0; kk < K; ++kk) {
            const int d = 1 << (kk >> 3);
            const int o = kk & 7;
            const int off = (HALO + OFFY(o) * d) * LW + (HALO + OFFX(o) * d); // >= 0
            acc = fmaf(wk[kk], base[off], acc);
        }
        dstb[(size_t)c * HW + p] = acc;
        __syncthreads();                           // reads done before buf is overwritten
    }
}

extern "C" void kernel_launch(void* const* d_in, const int* in_sizes, int n_in,
                              void* d_out, int out_size, void* d_ws, size_t ws_size,
                              hipStream_t stream) {
    (void)in_sizes; (void)n_in; (void)out_size; (void)ws_size;
    const float* imgs  = (const float*)d_in[0];   // [8,3,128,128] f32
    const float* masks = (const float*)d_in[1];   // [8,21,128,128] f32
    float* out = (float*)d_out;                   // [8,21,128,128] f32

    char* ws = (char*)d_ws;
    float* posAdd = (float*)ws;                                   // 32 f32
    float* affbuf = (float*)(ws + 256);                           // 8*16384*32 f32 (16.78 MB)
    float* bufA   = (float*)(ws + 256 + (size_t)NB * HW * K * 4); // 8*21*16384 f32 (11 MB)

    pos_kernel<<<1, 32, 0, stream>>>(posAdd);
    aff_kernel<<<dim3((H / TILE) * (W / TILE), NB), 256, 0, stream>>>(imgs, posAdd, affbuf);

    // ping-pong so iteration 10 (even) lands in d_out
    const float* src = masks;
    for (int i = 1; i <= NUM_ITER; ++i) {
        float* dst = (i & 1) ? bufA : out;
        iter_kernel<<<dim3((H / TILE) * (W / TILE), NB), 256, 0, stream>>>(src, affbuf, dst);
        src = dst;
    }
}